// RGCNConv_84439057039590
// MI455X (gfx1250) — compile-verified
//
#include <hip/hip_runtime.h>
#include <hip/hip_bf16.h>

typedef __attribute__((ext_vector_type(2))) float v2f;
typedef __attribute__((ext_vector_type(8))) float v8f;

#define NNODES 50000
#define NEDGES 600000
#define DIM    128
#define NREL   32
#define NBASES 4
#define BN_EPS 1e-5f

// 32-edge tiles: max tiles = sum_r ceil(cnt_r/32) <= ceil((E + 32*31)/32)
#define TILE_CAP 18784
#define PERM_CAP (TILE_CAP * 32)
#define LDS_STRIDE 132  // 128 + 4 pad: banks = (4*row + k) % 64, conflict-free

// packed-B layout: 33 matrices (0 = W_loop, 1+r = relation r), each
// 8 col-tiles x 32 k-steps x 32 lanes of float2 -> 8192 v2f per matrix.
#define FRAGS_PER_MAT 8192

__device__ __forceinline__ v8f wmma_f32_16x16x4(v2f a, v2f b, v8f c) {
    return __builtin_amdgcn_wmma_f32_16x16x4_f32(false, a, false, b, (short)0, c,
                                                 false, false);
}

// ---------------- workspace init (ws reused across graph replays) -----------
__global__ void k_init(int* __restrict__ perm, int* __restrict__ tileRel,
                       int* __restrict__ deg, int* __restrict__ counts,
                       int* __restrict__ cursors, float* __restrict__ stats) {
    int i = blockIdx.x * blockDim.x + threadIdx.x;
    if (i < PERM_CAP) perm[i] = -1;
    if (i < TILE_CAP) tileRel[i] = -1;
    if (i < NNODES)   deg[i] = 0;
    if (i < NREL)   { counts[i] = 0; cursors[i] = 0; }
    if (i < 2 * DIM)  stats[i] = 0.0f;   // sum | sumsq
}

// ---- pack W_loop and W[r]=sum_b coeff[r,b]*bases[b] into WMMA B fragments --
// pB[m*8192 + w*1024 + j*32 + lane] = { W[kk][col], W[kk+1][col] }
//   kk = j*4 + (lane>>4)*2, col = w*16 + (lane&15)
__global__ void k_pack(const float* __restrict__ coeff,
                       const float* __restrict__ bases,
                       const float* __restrict__ W_loop,
                       v2f* __restrict__ pB) {
    int i = blockIdx.x * blockDim.x + threadIdx.x;
    if (i >= (NREL + 1) * FRAGS_PER_MAT) return;
    int m = i >> 13;                 // matrix id
    int rem = i & 8191;
    int w = rem >> 10;
    int j = (rem >> 5) & 31;
    int l = rem & 31;
    int kk = j * 4 + (l >> 4) * 2;
    int col = w * 16 + (l & 15);
    int i0 = kk * DIM + col;
    int i1 = (kk + 1) * DIM + col;
    float x0, x1;
    if (m == 0) {
        x0 = W_loop[i0];
        x1 = W_loop[i1];
    } else {
        int r = m - 1;
        x0 = 0.0f; x1 = 0.0f;
        #pragma unroll
        for (int b = 0; b < NBASES; ++b) {
            float cc = coeff[r * NBASES + b];
            x0 = fmaf(cc, bases[b * DIM * DIM + i0], x0);
            x1 = fmaf(cc, bases[b * DIM * DIM + i1], x1);
        }
    }
    v2f v; v.x = x0; v.y = x1;
    pB[i] = v;
}

// ---------------- histogram of relations + in-degrees -----------------------
__global__ void k_count(const int* __restrict__ edge_index,
                        const int* __restrict__ edge_type,
                        int* __restrict__ counts, int* __restrict__ deg) {
    int e = blockIdx.x * blockDim.x + threadIdx.x;
    if (e >= NEDGES) return;
    atomicAdd(&counts[edge_type[e]], 1);
    atomicAdd(&deg[edge_index[NEDGES + e]], 1);
}

// ---------------- exclusive prefix, padded to 32 ----------------------------
__global__ void k_prefix(const int* __restrict__ counts, int* __restrict__ offsets) {
    if (threadIdx.x != 0 || blockIdx.x != 0) return;
    int off = 0;
    for (int r = 0; r < NREL; ++r) {
        offsets[r] = off;
        off += (counts[r] + 31) & ~31;
    }
    offsets[NREL] = off;
}

__global__ void k_tilerel(const int* __restrict__ counts,
                          const int* __restrict__ offsets,
                          int* __restrict__ tileRel) {
    int r = threadIdx.x;
    if (r >= NREL) return;
    int base = offsets[r] >> 5;
    int nt = (counts[r] + 31) >> 5;
    for (int i = 0; i < nt; ++i) tileRel[base + i] = r;
}

__global__ void k_scatter(const int* __restrict__ edge_type,
                          const int* __restrict__ offsets,
                          int* __restrict__ cursors, int* __restrict__ perm) {
    int e = blockIdx.x * blockDim.x + threadIdx.x;
    if (e >= NEDGES) return;
    int t = edge_type[e];
    int pos = atomicAdd(&cursors[t], 1);
    perm[offsets[t] + pos] = e;
}

// ---------------- self-loop GEMM: out = emb @ W_loop + b_loop ---------------
// block = 256 threads = 8 waves; block -> 32 rows x 128 cols; wave -> 16 cols,
// two 16-row accumulators sharing each B fragment.
__global__ void k_selfloop(const float* __restrict__ emb,
                           const v2f* __restrict__ pB,
                           const float* __restrict__ bl,
                           float* __restrict__ out) {
    __shared__ float sA[32 * LDS_STRIDE];
    const int tid = threadIdx.x;
    const int row0 = blockIdx.x * 32;

    #pragma unroll
    for (int i = 0; i < 4; ++i) {
        int u = tid + i * 256;          // 1024 float4 slots
        int e = u >> 5, c4 = u & 31;
        int row = row0 + e;
        float4 v = make_float4(0.f, 0.f, 0.f, 0.f);
        if (row < NNODES) v = *(const float4*)&emb[row * DIM + c4 * 4];
        *(float4*)&sA[e * LDS_STRIDE + c4 * 4] = v;
    }
    __syncthreads();

    const int wave = tid >> 5, lane = tid & 31;
    const int half = lane >> 4;              // K-sub-block select
    const int l15 = lane & 15;               // A row / B-C-D col
    const int col = wave * 16 + l15;
    const v2f* __restrict__ Bw = pB + wave * 1024 + lane;   // matrix m = 0

    v8f c0 = {}, c1 = {};
    #pragma unroll
    for (int j = 0; j < 32; ++j) {
        int kk = j * 4 + half * 2;
        v2f b = Bw[j * 32];                               // coalesced b64
        v2f a0 = *(const v2f*)&sA[l15 * LDS_STRIDE + kk];
        v2f a1 = *(const v2f*)&sA[(16 + l15) * LDS_STRIDE + kk];
        c0 = wmma_f32_16x16x4(a0, b, c0);
        c1 = wmma_f32_16x16x4(a1, b, c1);
    }

    float bias = bl[col];
    #pragma unroll
    for (int v = 0; v < 8; ++v) {
        int m = half * 8 + v;
        int r0 = row0 + m, r1 = row0 + 16 + m;
        if (r0 < NNODES) out[r0 * DIM + col] = c0[v] + bias;
        if (r1 < NNODES) out[r1 * DIM + col] = c1[v] + bias;
    }
}

// ---------------- per-edge messages (relation-pure 32-edge tiles) -----------
__global__ void k_edges(const float* __restrict__ emb,
                        const v2f* __restrict__ pB,
                        const int* __restrict__ edge_index,
                        const int* __restrict__ perm,
                        const int* __restrict__ tileRel,
                        float* __restrict__ out) {
    const int t = blockIdx.x;
    const int r = tileRel[t];
    if (r < 0) return;

    __shared__ float sA[32 * LDS_STRIDE];
    __shared__ int sSrc[32], sDst[32];
    const int tid = threadIdx.x;

    if (tid < 32) {
        int p = perm[t * 32 + tid];
        sSrc[tid] = (p >= 0) ? edge_index[p] : -1;
        sDst[tid] = (p >= 0) ? edge_index[NEDGES + p] : -1;
    }
    __syncthreads();

    // gather 32 source rows into LDS
    #pragma unroll
    for (int i = 0; i < 4; ++i) {
        int u = tid + i * 256;
        int e = u >> 5, c4 = u & 31;
        int srow = sSrc[e];
        float4 v = make_float4(0.f, 0.f, 0.f, 0.f);
        if (srow >= 0) v = *(const float4*)&emb[srow * DIM + c4 * 4];
        *(float4*)&sA[e * LDS_STRIDE + c4 * 4] = v;
    }
    __syncthreads();

    const int wave = tid >> 5, lane = tid & 31;
    const int half = lane >> 4;
    const int l15 = lane & 15;
    const int col = wave * 16 + l15;
    const v2f* __restrict__ Bw = pB + (r + 1) * FRAGS_PER_MAT + wave * 1024 + lane;

    v8f c0 = {}, c1 = {};
    #pragma unroll
    for (int j = 0; j < 32; ++j) {
        int kk = j * 4 + half * 2;
        v2f b = Bw[j * 32];                               // coalesced b64
        v2f a0 = *(const v2f*)&sA[l15 * LDS_STRIDE + kk];
        v2f a1 = *(const v2f*)&sA[(16 + l15) * LDS_STRIDE + kk];
        c0 = wmma_f32_16x16x4(a0, b, c0);
        c1 = wmma_f32_16x16x4(a1, b, c1);
    }

    // segment-sum scatter
    #pragma unroll
    for (int v = 0; v < 8; ++v) {
        int m = half * 8 + v;
        int d0 = sDst[m], d1 = sDst[16 + m];
        if (d0 >= 0) atomicAdd(&out[d0 * DIM + col], c0[v]);
        if (d1 >= 0) atomicAdd(&out[d1 * DIM + col], c1[v]);
    }
}

// ---------------- degree normalize + per-channel partial sums ---------------
__global__ void k_degnorm_stats(float* __restrict__ out, const int* __restrict__ deg,
                                float* __restrict__ stats) {
    const int col = threadIdx.x;            // blockDim = 128
    const int n0 = blockIdx.x * 128;
    float s = 0.f, s2 = 0.f;
    for (int j = 0; j < 128; ++j) {
        int n = n0 + j;
        if (n >= NNODES) break;
        float rdeg = 1.0f / fmaxf((float)deg[n], 1.0f);
        float v = out[n * DIM + col] * rdeg;
        out[n * DIM + col] = v;
        s += v;
        s2 += v * v;
    }
    atomicAdd(&stats[col], s);
    atomicAdd(&stats[DIM + col], s2);
}

__global__ void k_finalize(const float* __restrict__ stats,
                           float* __restrict__ mv) {
    int c = threadIdx.x;
    if (c >= DIM) return;
    float inv_n = 1.0f / (float)NNODES;
    float m = stats[c] * inv_n;
    float var = stats[DIM + c] * inv_n - m * m;
    mv[c] = m;
    mv[DIM + c] = __frsqrt_rn(fmaxf(var, 0.0f) + BN_EPS);
}

__global__ void k_bn_relu(float* __restrict__ out, const float* __restrict__ mv,
                          const float* __restrict__ gamma,
                          const float* __restrict__ beta) {
    int i = blockIdx.x * blockDim.x + threadIdx.x;
    if (i >= NNODES * DIM) return;
    int c = i & (DIM - 1);
    float y = (out[i] - mv[c]) * mv[DIM + c] * gamma[c] + beta[c];
    out[i] = fmaxf(y, 0.0f);
}

// ---------------------------------------------------------------------------
extern "C" void kernel_launch(void* const* d_in, const int* in_sizes, int n_in,
                              void* d_out, int out_size, void* d_ws, size_t ws_size,
                              hipStream_t stream) {
    const float* emb     = (const float*)d_in[0];
    const float* bases   = (const float*)d_in[1];
    const float* coeff   = (const float*)d_in[2];
    const float* W_loop  = (const float*)d_in[3];
    const float* b_loop  = (const float*)d_in[4];
    const float* gamma   = (const float*)d_in[5];
    const float* beta    = (const float*)d_in[6];
    const int*   edge_ix = (const int*)d_in[7];
    const int*   edge_ty = (const int*)d_in[8];
    float* out = (float*)d_out;

    // workspace carve-up (all 16B-aligned by construction)
    char* p = (char*)d_ws;
    v2f*   pB      = (v2f*)p;   p += (size_t)(NREL + 1) * FRAGS_PER_MAT * 8; // 2.1 MB
    int*   counts  = (int*)p;   p += 32 * 4;
    int*   cursors = (int*)p;   p += 32 * 4;
    int*   offsets = (int*)p;   p += 36 * 4;
    int*   tileRel = (int*)p;   p += (size_t)TILE_CAP * 4;
    int*   perm    = (int*)p;   p += (size_t)PERM_CAP * 4;
    int*   deg     = (int*)p;   p += (size_t)NNODES * 4;
    float* stats   = (float*)p; p += 2 * DIM * 4;   // sum | sumsq
    float* mv      = (float*)p; p += 2 * DIM * 4;   // mean | invstd

    k_init<<<(PERM_CAP + 255) / 256, 256, 0, stream>>>(perm, tileRel, deg,
                                                       counts, cursors, stats);
    k_pack<<<((NREL + 1) * FRAGS_PER_MAT + 255) / 256, 256, 0, stream>>>(
        coeff, bases, W_loop, pB);
    k_count<<<(NEDGES + 255) / 256, 256, 0, stream>>>(edge_ix, edge_ty, counts, deg);
    k_prefix<<<1, 32, 0, stream>>>(counts, offsets);
    k_tilerel<<<1, 32, 0, stream>>>(counts, offsets, tileRel);
    k_scatter<<<(NEDGES + 255) / 256, 256, 0, stream>>>(edge_ty, offsets, cursors, perm);
    k_selfloop<<<(NNODES + 31) / 32, 256, 0, stream>>>(emb, pB, b_loop, out);
    k_edges<<<TILE_CAP, 256, 0, stream>>>(emb, pB, edge_ix, perm, tileRel, out);
    k_degnorm_stats<<<(NNODES + 127) / 128, 128, 0, stream>>>(out, deg, stats);
    k_finalize<<<1, 128, 0, stream>>>(stats, mv);
    k_bn_relu<<<(NNODES * DIM + 255) / 256, 256, 0, stream>>>(out, mv, gamma, beta);
}